// GCN_75230647157565
// MI455X (gfx1250) — compile-verified
//
#include <hip/hip_runtime.h>

#define N_NODES 100000
#define N_EDGES 1600000

typedef __attribute__((ext_vector_type(2))) float v2f;
typedef __attribute__((ext_vector_type(8))) float v8f;

// ---------------------------------------------------------------------------
// fp32 atomic add that lowers to global_atomic_add_f32 (no CAS loop)
// ---------------------------------------------------------------------------
__device__ __forceinline__ void f32_atomic_add(float* p, float v) {
    unsafeAtomicAdd(p, v);
}

// ---------------------------------------------------------------------------
// zero a buffer (float4 granularity) -- only used for the degree pass
// ---------------------------------------------------------------------------
__global__ void zero4_kernel(float4* __restrict__ p, int n4) {
    int t = blockIdx.x * blockDim.x + threadIdx.x;
    if (t < n4) p[t] = make_float4(0.f, 0.f, 0.f, 0.f);
}

// ---------------------------------------------------------------------------
// degree (self loop added in dinv_kernel): deg[dst[e]] += 1
// ---------------------------------------------------------------------------
__global__ void degree_kernel(const int* __restrict__ dst, float* __restrict__ deg) {
    int e = blockIdx.x * blockDim.x + threadIdx.x;
    if (e < N_EDGES) f32_atomic_add(&deg[dst[e]], 1.0f);
}

__global__ void dinv_kernel(float* __restrict__ deg_dinv) {
    int i = blockIdx.x * blockDim.x + threadIdx.x;
    if (i < N_NODES) deg_dinv[i] = rsqrtf(deg_dinv[i] + 1.0f);
}

// ---------------------------------------------------------------------------
// Y[m, n] = dinv[m] * (H @ W)[m, n]   via V_WMMA_F32_16X16X4_F32
// Also seeds AGG with the same value (self-loop contribution), eliminating a
// separate zero pass and halving combine's read traffic.
// grid.x = N_NODES/16 (m-tiles); blockDim.x = (COUT/16)*32, wave w -> n-tile w
// Exact tiling; all lanes active -> EXEC all ones for WMMA.
// ---------------------------------------------------------------------------
template <int K, int COUT>
__global__ void gemm_dinv_wmma(const float* __restrict__ H,
                               const float* __restrict__ W,
                               const float* __restrict__ dinv,
                               float* __restrict__ Y,
                               float* __restrict__ AGG) {
    const int lane   = threadIdx.x & 31;
    const int n_tile = threadIdx.x >> 5;          // one 16-wide N tile per wave
    const int m_tile = blockIdx.x;

    const int row = m_tile * 16 + (lane & 15);    // A row (lanes 0-15 / 16-31 same rows)
    const int col = n_tile * 16 + (lane & 15);    // B/D column
    const int kh  = (lane >> 4) << 1;             // lanes 0-15: K+{0,1}; 16-31: K+{2,3}

    const float* hrow = H + row * K + kh;

    v8f acc = {};
#pragma unroll
    for (int kk = 0; kk < K; kk += 4) {
        v2f a = *reinterpret_cast<const v2f*>(hrow + kk);   // A[row, kk+kh .. +1]
        v2f b;
        b.x = W[(kk + kh) * COUT + col];                    // B[kk+kh,   col]
        b.y = W[(kk + kh + 1) * COUT + col];                // B[kk+kh+1, col]
        acc = __builtin_amdgcn_wmma_f32_16x16x4_f32(
            /*neg_a=*/false, a, /*neg_b=*/false, b,
            /*c_mod=*/(short)0, acc, /*reuse_a=*/false, /*reuse_b=*/false);
    }

    // D layout: VGPR r -> M = 16*m_tile + r + (lane>=16 ? 8 : 0), N = col
    const int mbase = m_tile * 16 + ((lane >> 4) << 3);
#pragma unroll
    for (int r = 0; r < 8; ++r) {
        const int m   = mbase + r;
        const float v = dinv[m] * acc[r];
        const int idx = m * COUT + col;
        Y[idx]   = v;   // immutable copy for the edge gather
        AGG[idx] = v;   // seed aggregation with the self-loop term
    }
}

// ---------------------------------------------------------------------------
// AGG[dst[e], :] += Y[src[e], :]
// One thread per (edge, 4-float chunk): a wave covers one full 128-ch edge row
// -> 512B contiguous gather + contiguous fp32 atomics resolving in L2.
// ---------------------------------------------------------------------------
template <int COUT>
__global__ void scatter_kernel(const int* __restrict__ src,
                               const int* __restrict__ dst,
                               const float* __restrict__ Y,
                               float* __restrict__ AGG) {
    constexpr int CH = COUT / 4;                  // float4 chunks per row
    int t = blockIdx.x * blockDim.x + threadIdx.x;
    if (t >= N_EDGES * CH) return;
    int e  = t / CH;                              // CH is a power of two -> shift
    int c4 = (t - e * CH) * 4;
    int s = src[e];
    int d = dst[e];
    const float4 v = *reinterpret_cast<const float4*>(Y + s * COUT + c4);
    float* o = AGG + d * COUT + c4;
    f32_atomic_add(o + 0, v.x);
    f32_atomic_add(o + 1, v.y);
    f32_atomic_add(o + 2, v.z);
    f32_atomic_add(o + 3, v.w);
}

// ---------------------------------------------------------------------------
// OUT[i, c] = (relu)( dinv[i] * AGG[i,c] + bias[c] )
// (AGG already contains self-loop + neighbor sum, all pre-scaled by dinv[src])
// ---------------------------------------------------------------------------
template <int COUT, bool RELU>
__global__ void combine_kernel(const float* __restrict__ AGG,
                               const float* __restrict__ dinv,
                               const float* __restrict__ bias,
                               float* __restrict__ OUT) {
    constexpr int SH = (COUT == 128) ? 7 : 6;
    int t = blockIdx.x * blockDim.x + threadIdx.x;
    if (t >= N_NODES * COUT) return;
    int i = t >> SH;
    int c = t & (COUT - 1);
    float v = dinv[i] * AGG[t] + bias[c];
    if (RELU) v = fmaxf(v, 0.0f);
    OUT[t] = v;
}

// ---------------------------------------------------------------------------
// launch
// ---------------------------------------------------------------------------
extern "C" void kernel_launch(void* const* d_in, const int* in_sizes, int n_in,
                              void* d_out, int out_size, void* d_ws, size_t ws_size,
                              hipStream_t stream) {
    const float* x  = (const float*)d_in[0];       // [N, 64]
    const int*   ei = (const int*)d_in[1];         // [2, E]
    const float* W1 = (const float*)d_in[2];       // [64, 128]
    const float* b1 = (const float*)d_in[3];
    const float* W2 = (const float*)d_in[4];       // [128, 128]
    const float* b2 = (const float*)d_in[5];
    const float* W3 = (const float*)d_in[6];       // [128, 64]
    const float* b3 = (const float*)d_in[7];
    float*       out = (float*)d_out;              // [N, 64]

    const int* src = ei;
    const int* dst = ei + N_EDGES;

    // workspace carve-up (aligned to 512B)
    char*  ws   = (char*)d_ws;
    size_t off  = 0;
    float* dinv = (float*)(ws + off); off += ((size_t)N_NODES * 4 + 511) & ~511ull;
    float* y    = (float*)(ws + off); off += (size_t)N_NODES * 128 * 4;
    float* agg  = (float*)(ws + off); off += (size_t)N_NODES * 128 * 4;
    float* h    = (float*)(ws + off);

    const int B = 256;
    const int M_TILES = N_NODES / 16;              // 6250, exact

    // ---- degree -> dinv (in place) ----
    zero4_kernel<<<(N_NODES / 4 + B - 1) / B, B, 0, stream>>>((float4*)dinv, N_NODES / 4);
    degree_kernel<<<(N_EDGES + B - 1) / B, B, 0, stream>>>(dst, dinv);
    dinv_kernel<<<(N_NODES + B - 1) / B, B, 0, stream>>>(dinv);

    // ---- layer 1: 64 -> 128, ReLU ----
    gemm_dinv_wmma<64, 128><<<M_TILES, 256, 0, stream>>>(x, W1, dinv, y, agg);
    scatter_kernel<128><<<(N_EDGES * 32 + B - 1) / B, B, 0, stream>>>(src, dst, y, agg);
    combine_kernel<128, true><<<(N_NODES * 128 + B - 1) / B, B, 0, stream>>>(agg, dinv, b1, h);

    // ---- layer 2: 128 -> 128, ReLU ----
    gemm_dinv_wmma<128, 128><<<M_TILES, 256, 0, stream>>>(h, W2, dinv, y, agg);
    scatter_kernel<128><<<(N_EDGES * 32 + B - 1) / B, B, 0, stream>>>(src, dst, y, agg);
    combine_kernel<128, true><<<(N_NODES * 128 + B - 1) / B, B, 0, stream>>>(agg, dinv, b2, h);

    // ---- layer 3: 128 -> 64, no activation ----
    gemm_dinv_wmma<128, 64><<<M_TILES, 128, 0, stream>>>(h, W3, dinv, y, agg);
    scatter_kernel<64><<<(N_EDGES * 16 + B - 1) / B, B, 0, stream>>>(src, dst, y, agg);
    combine_kernel<64, false><<<(N_NODES * 64 + B - 1) / B, B, 0, stream>>>(agg, dinv, b3, out);
}